// SIBlocks_17308718203258
// MI455X (gfx1250) — compile-verified
//
#include <hip/hip_runtime.h>
#include <math.h>

typedef __attribute__((ext_vector_type(2))) float v2f;
typedef __attribute__((ext_vector_type(8))) float v8f;

#define NPTS   4096
#define HGRID  64
#define CDIM   64
#define BDIM   4
#define KNBR   32
#define KNOTS  32
#define HID    128
#define NPART  16

__device__ __forceinline__ float lin63(int i) { return (float)i / 63.0f; }

// ---------------------------------------------------------------------------
// Kernel 1: per-point 32-NN (full scan, JAX top_k tie semantics) + raw psi/phi
// + deterministic per-block partial sums of |psi|, |phi|.
// ---------------------------------------------------------------------------
__global__ void knn_psi_phi_kernel(const float* __restrict__ Wi,
                                   const float* __restrict__ Wj,
                                   const float* __restrict__ pw,
                                   const float* __restrict__ Smx,
                                   const float* __restrict__ Smy,
                                   int*   __restrict__ idx_out,
                                   float* __restrict__ psi_out,
                                   float* __restrict__ phi_out,
                                   float* __restrict__ part_psi,
                                   float* __restrict__ part_phi) {
  const int tid = threadIdx.x;
  const int n   = blockIdx.x * blockDim.x + tid;   // 0..4095

  float bd[KNBR];
  int   bi[KNBR];
  #pragma unroll
  for (int t = 0; t < KNBR; ++t) { bd[t] = INFINITY; bi[t] = 0; }

  const float cx = lin63(n >> 6);
  const float cy = lin63(n & 63);

  // full scan in ascending candidate index => stable tie-break (lower idx wins)
  for (int cand = 0; cand < NPTS; ++cand) {
    float dx = cx - lin63(cand >> 6);
    float dy = cy - lin63(cand & 63);
    float d  = sqrtf(dx * dx + dy * dy);
    if (d <= 0.2f && d < bd[KNBR - 1]) {
      int p = KNBR - 1;
      while (p > 0 && d < bd[p - 1]) {     // strict '<': ties stay behind earlier idx
        bd[p] = bd[p - 1]; bi[p] = bi[p - 1]; --p;
      }
      bd[p] = d; bi[p] = cand;
    }
  }

  float sum_psi = 0.0f, sum_phi = 0.0f;
  for (int k = 0; k < KNBR; ++k) {
    const int j = bi[k];
    const float cjx = lin63(j >> 6);
    const float cjy = lin63(j & 63);
    const float rx = cx - cjx;
    const float ry = cy - cjy;

    // hat-basis "bspline" with 32 knots on [0,1]
    float px = 0.0f, py = 0.0f;
    #pragma unroll
    for (int m = 0; m < KNOTS; ++m) {
      const float kn  = (float)m / 31.0f;
      const float dxm = fabsf(rx - kn);
      const float dym = fabsf(ry - kn);
      px += (dxm < 1.0f ? 1.0f - dxm : 0.0f) * Smx[m];
      py += (dym < 1.0f ? 1.0f - dym : 0.0f) * Smy[m];
    }
    const float psi = px * py;

    // phi[n,k] = sum_c (coords_i @ Wi^T)[c] * (coords_j @ Wj^T)[c] * pw[c]
    float acc = 0.0f;
    for (int c = 0; c < CDIM; ++c) {
      const float ie = cx * Wi[2 * c] + cy * Wi[2 * c + 1];
      const float je = cjx * Wj[2 * c] + cjy * Wj[2 * c + 1];
      acc += ie * je * pw[c];
    }

    idx_out[n * KNBR + k] = j;
    psi_out[n * KNBR + k] = psi;
    phi_out[n * KNBR + k] = acc;
    sum_psi += fabsf(psi);
    sum_phi += fabsf(acc);
  }

  // fixed-order block reduction (deterministic)
  __shared__ float sps[256], sph[256];
  sps[tid] = sum_psi; sph[tid] = sum_phi;
  for (int s = 128; s > 0; s >>= 1) {
    __syncthreads();
    if (tid < s) { sps[tid] += sps[tid + s]; sph[tid] += sph[tid + s]; }
  }
  if (tid == 0) { part_psi[blockIdx.x] = sps[0]; part_phi[blockIdx.x] = sph[0]; }
}

// ---------------------------------------------------------------------------
// Kernel 2: fold partials in fixed order -> normalization scales.
// ---------------------------------------------------------------------------
__global__ void finalize_scales_kernel(const float* __restrict__ part_psi,
                                       const float* __restrict__ part_phi,
                                       float* __restrict__ scales) {
  if (threadIdx.x == 0 && blockIdx.x == 0) {
    float sp = 0.0f, sf = 0.0f;
    for (int i = 0; i < NPART; ++i) { sp += part_psi[i]; sf += part_phi[i]; }
    const float inv = 1.0f / (float)(NPTS * KNBR);
    scales[0] = 1.0f / (sp * inv + 1e-6f);
    scales[1] = 1.0f / (sf * inv + 1e-6f);
  }
}

// ---------------------------------------------------------------------------
// Kernel 3: fused 2-layer MLP with f32 WMMA (16x16x4). One wave per 16 rows.
//   A frag layout (ISA 32-bit A 16x4): lane%16 = M row; VGPR v holds
//   K = (lane/16)*2 + v within each k-chunk of 4.
//   C/D layout: lane%16 = N col; VGPR v holds M = v + 8*(lane/16).
// ---------------------------------------------------------------------------
__global__ void mlp_wmma_kernel(const float* __restrict__ x,
                                const float* __restrict__ W1,
                                const float* __restrict__ b1,
                                const float* __restrict__ W2,
                                const float* __restrict__ b2,
                                float* __restrict__ out) {
  const int lane  = threadIdx.x;          // 0..31
  const int lrow  = lane & 15;
  const int lhalf = lane >> 4;            // 0 or 1
  const long r0   = (long)blockIdx.x * 16;

  __shared__ float hlds[16 * 132];        // 16 rows x 128 cols, padded stride

  // load A fragments for layer 1 (row = r0+lrow, K = 64 in 16 chunks of 4)
  v2f a[16];
  const float* xrow = x + (r0 + lrow) * CDIM;
  #pragma unroll
  for (int kc = 0; kc < 16; ++kc) {
    a[kc] = *(const v2f*)(xrow + kc * 4 + lhalf * 2);
  }

  // ---- layer 1: h = relu(x @ W1^T + b1), 8 col-blocks of 16 ----
  #pragma unroll
  for (int cb = 0; cb < 8; ++cb) {
    v8f acc = {};
    #pragma unroll
    for (int kc = 0; kc < 16; ++kc) {
      const int kb = kc * 4 + lhalf * 2;
      v2f bf = *(const v2f*)(W1 + (cb * 16 + lrow) * CDIM + kb);
      acc = __builtin_amdgcn_wmma_f32_16x16x4_f32(
          false, a[kc], false, bf, (short)0, acc, false, false);
    }
    const float bias = b1[cb * 16 + lrow];
    #pragma unroll
    for (int v = 0; v < 8; ++v) {
      float hv = acc[v] + bias;
      hv = hv > 0.0f ? hv : 0.0f;
      hlds[(v + lhalf * 8) * 132 + cb * 16 + lrow] = hv;
    }
  }
  __syncthreads();

  // ---- layer 2: pointwise = h @ W2^T + b2, 4 col-blocks of 16, K=128 ----
  #pragma unroll
  for (int cb = 0; cb < 4; ++cb) {
    v8f acc = {};
    #pragma unroll
    for (int kc = 0; kc < 32; ++kc) {
      const int kb = kc * 4 + lhalf * 2;
      v2f af = { hlds[lrow * 132 + kb], hlds[lrow * 132 + kb + 1] };
      v2f bf = *(const v2f*)(W2 + (cb * 16 + lrow) * HID + kb);
      acc = __builtin_amdgcn_wmma_f32_16x16x4_f32(
          false, af, false, bf, (short)0, acc, false, false);
    }
    const float bias = b2[cb * 16 + lrow];
    #pragma unroll
    for (int v = 0; v < 8; ++v) {
      const int m = v + lhalf * 8;
      out[(r0 + m) * CDIM + cb * 16 + lrow] = acc[v] + bias;
    }
  }
}

// ---------------------------------------------------------------------------
// Kernel 4: neighbor gather-reduce, added on top of the pointwise result.
// Block per (b,n), thread per channel c.
// ---------------------------------------------------------------------------
__global__ void gather_add_kernel(const float* __restrict__ x,
                                  const int*   __restrict__ idx,
                                  const float* __restrict__ psi_raw,
                                  const float* __restrict__ phi_raw,
                                  const float* __restrict__ scales,
                                  float* __restrict__ out) {
  const int bn = blockIdx.x;            // 0 .. B*N-1
  const int b  = bn >> 12;
  const int n  = bn & (NPTS - 1);
  const int c  = threadIdx.x;           // 0..63
  const float sp = scales[0];
  const float sf = scales[1];

  const float* xb = x + (long)b * NPTS * CDIM;
  float acc = 0.0f;
  for (int k = 0; k < KNBR; ++k) {
    const int   j = idx[n * KNBR + k];
    const float w = (psi_raw[n * KNBR + k] * sp) * (phi_raw[n * KNBR + k] * sf);
    acc += w * xb[(long)j * CDIM + c];
  }
  out[(long)bn * CDIM + c] += acc * (1.0f / (float)KNBR);
}

// ---------------------------------------------------------------------------
extern "C" void kernel_launch(void* const* d_in, const int* in_sizes, int n_in,
                              void* d_out, int out_size, void* d_ws, size_t ws_size,
                              hipStream_t stream) {
  const float* x   = (const float*)d_in[0];
  const float* Wi  = (const float*)d_in[1];
  const float* Wj  = (const float*)d_in[2];
  const float* pw  = (const float*)d_in[3];
  // d_in[4..7] = h_w1,h_b1,h_w2,h_b2 -> dead code in reference
  const float* Smx = (const float*)d_in[8];
  const float* Smy = (const float*)d_in[9];
  const float* W1  = (const float*)d_in[10];
  const float* b1  = (const float*)d_in[11];
  const float* W2  = (const float*)d_in[12];
  const float* b2  = (const float*)d_in[13];
  float* out = (float*)d_out;

  char*  ws       = (char*)d_ws;
  int*   idx      = (int*)ws;                                   // N*K ints
  float* psi_raw  = (float*)(ws + sizeof(int) * NPTS * KNBR);   // N*K floats
  float* phi_raw  = psi_raw + NPTS * KNBR;                      // N*K floats
  float* part_psi = phi_raw + NPTS * KNBR;                      // NPART floats
  float* part_phi = part_psi + NPART;                           // NPART floats
  float* scales   = part_phi + NPART;                           // 2 floats

  knn_psi_phi_kernel<<<NPART, 256, 0, stream>>>(Wi, Wj, pw, Smx, Smy,
                                                idx, psi_raw, phi_raw,
                                                part_psi, part_phi);
  finalize_scales_kernel<<<1, 32, 0, stream>>>(part_psi, part_phi, scales);
  mlp_wmma_kernel<<<(BDIM * NPTS) / 16, 32, 0, stream>>>(x, W1, b1, W2, b2, out);
  gather_add_kernel<<<BDIM * NPTS, CDIM, 0, stream>>>(x, idx, psi_raw, phi_raw,
                                                      scales, out);
}